// GATLayer_24008867184975
// MI455X (gfx1250) — compile-verified
//
#include <hip/hip_runtime.h>
#include <limits.h>
#include <math.h>

#define N_NODES   50000
#define N_EDGES   800000
#define IN_F      128
#define OUT_F     16
#define HEADS     4
#define HF        64            // HEADS*OUT_F
#define NEG_SLOPE 0.2f

typedef __attribute__((ext_vector_type(2))) float v2f;
typedef __attribute__((ext_vector_type(8))) float v8f;

// ---------------------------------------------------------------------------
// Kernel 1: h = x @ W   (50000x128 @ 128x64, fp32)  via V_WMMA_F32_16X16X4_F32
// 625 blocks x 128 threads (4 waves); each block: 5 row-tiles of 16 nodes,
// each wave owns a 16-wide column strip. W staged once per block into LDS.
// ---------------------------------------------------------------------------
#define ROWT      5
#define XS_STRIDE 132   // 132 mod 64 = 4  -> 16 rows hit 16 distinct bank groups
#define WS_STRIDE 72    // 72  mod 64 = 8  -> khalf groups use disjoint banks

__global__ __launch_bounds__(128)
void gat_gemm(const float* __restrict__ x, const float* __restrict__ W,
              float* __restrict__ h) {
  __shared__ float Ws[IN_F * WS_STRIDE];
  __shared__ float Xs[16 * XS_STRIDE];

  const int tid   = threadIdx.x;
  const int wave  = tid >> 5;
  const int lane  = tid & 31;
  const int nrow  = lane & 15;   // M index (A) / N index (B, D)
  const int khalf = lane >> 4;   // 0 -> K{0,1}, 1 -> K{2,3}
  const int col0  = wave * 16;

  // stage W (128x64) into padded LDS
  for (int idx = tid; idx < IN_F * HF; idx += 128)
    Ws[(idx >> 6) * WS_STRIDE + (idx & 63)] = W[idx];

  for (int t = 0; t < ROWT; ++t) {
    const int node0 = blockIdx.x * (16 * ROWT) + t * 16;
    __syncthreads();                       // Ws ready (t==0) / Xs consumed (t>0)
    for (int idx = tid; idx < 16 * IN_F; idx += 128)
      Xs[(idx >> 7) * XS_STRIDE + (idx & 127)] = x[node0 * IN_F + idx];
    __syncthreads();

    v8f c = {0.f, 0.f, 0.f, 0.f, 0.f, 0.f, 0.f, 0.f};
    #pragma unroll
    for (int k = 0; k < IN_F; k += 4) {
      const int kk = k + 2 * khalf;
      v2f a = *reinterpret_cast<const v2f*>(&Xs[nrow * XS_STRIDE + kk]);
      v2f b;
      b.x = Ws[(kk    ) * WS_STRIDE + col0 + nrow];
      b.y = Ws[(kk + 1) * WS_STRIDE + col0 + nrow];
      c = __builtin_amdgcn_wmma_f32_16x16x4_f32(false, a, false, b,
                                                (short)0, c, false, false);
    }
    #pragma unroll
    for (int r = 0; r < 8; ++r) {
      const int m = r + khalf * 8;
      h[(node0 + m) * HF + col0 + nrow] = c[r];
    }
  }
}

// ---------------------------------------------------------------------------
// Kernel 2: init denom = 0, global-max key = INT_MIN
// ---------------------------------------------------------------------------
__global__ void gat_init(float* __restrict__ denom, int* __restrict__ gkey) {
  const int i = blockIdx.x * blockDim.x + threadIdx.x;
  if (i < N_NODES * HEADS) denom[i] = 0.f;
  if (i == 0) *gkey = INT_MIN;
}

// ---------------------------------------------------------------------------
// Kernel 3: per-node attention dots  s[n,h] = <h[n,h,:], a_src[h]>, same for t
// one thread per (node, head): 64B contiguous read of h
// ---------------------------------------------------------------------------
__global__ __launch_bounds__(256)
void gat_dots(const float* __restrict__ h, const float* __restrict__ a_src,
              const float* __restrict__ a_tgt,
              float* __restrict__ s, float* __restrict__ t) {
  const int i = blockIdx.x * blockDim.x + threadIdx.x;
  if (i >= N_NODES * HEADS) return;
  const int head = i & (HEADS - 1);
  const float* hp = h + (i >> 2) * HF + head * OUT_F;
  const float* as = a_src + head * OUT_F;
  const float* at = a_tgt + head * OUT_F;
  float ss = 0.f, tt = 0.f;
  #pragma unroll
  for (int f = 0; f < OUT_F; ++f) {
    const float v = hp[f];
    ss = fmaf(v, as[f], ss);
    tt = fmaf(v, at[f], tt);
  }
  s[i] = ss;
  t[i] = tt;
}

// order-preserving float <-> signed-int key (for atomicMax on float)
__device__ __forceinline__ int f2key(float f) {
  const int i = __float_as_int(f);
  return i >= 0 ? i : (i ^ 0x7FFFFFFF);
}
__device__ __forceinline__ float key2f(int k) {
  return __int_as_float(k >= 0 ? k : (k ^ 0x7FFFFFFF));
}

__device__ __forceinline__ float leaky(float e) {
  return e > 0.f ? e : NEG_SLOPE * e;
}

// ---------------------------------------------------------------------------
// Kernel 4: global max of leaky(e) over all edges x heads
// s/t tables (800KB each) are L2 resident -> gathers hit L2
// ---------------------------------------------------------------------------
__global__ __launch_bounds__(256)
void gat_edge_max(const int* __restrict__ ei, const float* __restrict__ s,
                  const float* __restrict__ t, int* __restrict__ gkey) {
  __shared__ float red[256];
  float m = -INFINITY;
  for (int e = blockIdx.x * blockDim.x + threadIdx.x; e < N_EDGES;
       e += gridDim.x * blockDim.x) {
    const int src = ei[e];
    const int tgt = ei[N_EDGES + e];
    const float4 sv = *reinterpret_cast<const float4*>(s + 4 * src);
    const float4 tv = *reinterpret_cast<const float4*>(t + 4 * tgt);
    m = fmaxf(m, leaky(sv.x + tv.x));
    m = fmaxf(m, leaky(sv.y + tv.y));
    m = fmaxf(m, leaky(sv.z + tv.z));
    m = fmaxf(m, leaky(sv.w + tv.w));
  }
  const int tid = threadIdx.x;
  red[tid] = m;
  __syncthreads();
  for (int off = 128; off > 0; off >>= 1) {
    if (tid < off) red[tid] = fmaxf(red[tid], red[tid + off]);
    __syncthreads();
  }
  if (tid == 0) atomicMax(gkey, f2key(red[0]));
}

// ---------------------------------------------------------------------------
// Kernel 5: e_exp = exp(leaky(e) - gmax); denom[tgt,h] += e_exp (L2 atomics)
// ---------------------------------------------------------------------------
__global__ __launch_bounds__(256)
void gat_edge_exp(const int* __restrict__ ei, const float* __restrict__ s,
                  const float* __restrict__ t, const int* __restrict__ gkey,
                  float* __restrict__ denom) {
  const int e = blockIdx.x * blockDim.x + threadIdx.x;
  if (e >= N_EDGES) return;
  const float gmax = key2f(*gkey);
  const int src = ei[e];
  const int tgt = ei[N_EDGES + e];
  const float4 sv = *reinterpret_cast<const float4*>(s + 4 * src);
  const float4 tv = *reinterpret_cast<const float4*>(t + 4 * tgt);
  float ex[HEADS];
  ex[0] = __expf(leaky(sv.x + tv.x) - gmax);
  ex[1] = __expf(leaky(sv.y + tv.y) - gmax);
  ex[2] = __expf(leaky(sv.z + tv.z) - gmax);
  ex[3] = __expf(leaky(sv.w + tv.w) - gmax);
  float* d = denom + tgt * HEADS;
  #pragma unroll
  for (int hh = 0; hh < HEADS; ++hh) atomicAdd(d + hh, ex[hh]);
}

// ---------------------------------------------------------------------------
// Kernel 6: out[n,c] = h[n,c] * denom/(denom+1e-16) + bias[c]   (in place)
// (since the reference aggregates h_tgt into tgt, h_prime = h * sum(alpha))
// ---------------------------------------------------------------------------
__global__ __launch_bounds__(256)
void gat_finalize(const float* __restrict__ denom, const float* __restrict__ bias,
                  float* __restrict__ out) {
  const int i = blockIdx.x * blockDim.x + threadIdx.x;
  if (i >= N_NODES * HF) return;
  const int c = i & (HF - 1);
  const int n = i >> 6;
  const float d = denom[n * HEADS + (c >> 4)];
  const float scale = d / (d + 1e-16f);
  out[i] = out[i] * scale + bias[c];
}

// ---------------------------------------------------------------------------
extern "C" void kernel_launch(void* const* d_in, const int* in_sizes, int n_in,
                              void* d_out, int out_size, void* d_ws, size_t ws_size,
                              hipStream_t stream) {
  const float* x     = (const float*)d_in[0];  // [N, 128]
  const int*   ei    = (const int*)  d_in[1];  // [2, E]
  const float* W     = (const float*)d_in[2];  // [128, 64]
  const float* a_src = (const float*)d_in[3];  // [1, 4, 16]
  const float* a_tgt = (const float*)d_in[4];  // [1, 4, 16]
  const float* bias  = (const float*)d_in[5];  // [64]
  float* out = (float*)d_out;                  // [N, 64] — also holds h

  float* s     = (float*)d_ws;                 // [N*4]
  float* t     = s + N_NODES * HEADS;          // [N*4]
  float* denom = t + N_NODES * HEADS;          // [N*4]
  int*   gkey  = (int*)(denom + N_NODES * HEADS);

  const int NH = N_NODES * HEADS;              // 200000

  gat_gemm<<<N_NODES / (16 * ROWT), 128, 0, stream>>>(x, W, out);
  gat_init<<<(NH + 255) / 256, 256, 0, stream>>>(denom, gkey);
  gat_dots<<<(NH + 255) / 256, 256, 0, stream>>>(out, a_src, a_tgt, s, t);
  gat_edge_max<<<512, 256, 0, stream>>>(ei, s, t, gkey);
  gat_edge_exp<<<(N_EDGES + 255) / 256, 256, 0, stream>>>(ei, s, t, gkey, denom);
  gat_finalize<<<(N_NODES * HF + 255) / 256, 256, 0, stream>>>(denom, bias, out);
}